// Net_link_evaulate_2190433321526
// MI455X (gfx1250) — compile-verified
//
#include <hip/hip_runtime.h>
#include <hip/hip_bf16.h>

typedef __attribute__((ext_vector_type(2))) float v2f;
typedef __attribute__((ext_vector_type(8))) float v8f;

// Padded pair-row stride: 80 float2 -> hi half-wave lands on banks 32..63
#define BSTRIDE 80

// ---------------------------------------------------------------------------
// GEMM: C[nRows x 64] = act(A[nRows x K]) @ B[K x 64]   (act = relu if RELU)
// One wave computes a 16-row x 64-col strip with V_WMMA_F32_16X16X4_F32.
// B is staged in LDS pre-paired along K: BsP[k2*BSTRIDE + n] = {B[2k2][n], B[2k2+1][n]}
// so each WMMA B operand is a single aligned ds_load_b64.
// A-layout (ISA 7.12.2, 32-bit A 16x4): lanes0-15 M=0..15 K={0,1},
//                                       lanes16-31 M=0..15 K={2,3}
// C-layout: VGPR r -> lanes0-15 (M=r, N=lane), lanes16-31 (M=r+8, N=lane-16)
// ---------------------------------------------------------------------------
template <int RELU>
__global__ void gemm_wmma_f32(const float* __restrict__ A,
                              const float* __restrict__ B,
                              float* __restrict__ C,
                              int nRows, int K) {
  __shared__ float2 BsP[64 * BSTRIDE];   // up to K=128 -> 64 pair-rows (40KB)
  const int pairRows = K >> 1;
  for (int j = threadIdx.x; j < pairRows * 64; j += blockDim.x) {
    const int k2 = j >> 6;
    const int nn = j & 63;
    BsP[k2 * BSTRIDE + nn] =
        make_float2(B[(2 * k2) * 64 + nn], B[(2 * k2 + 1) * 64 + nn]);
  }
  __syncthreads();

  const int lane = threadIdx.x & 31;
  const int wave = threadIdx.x >> 5;
  const int nStrips = nRows >> 4;                 // nRows % 16 == 0 (100000/16)
  const int strip = blockIdx.x * 8 + wave;        // 8 waves / block
  if (strip >= nStrips) return;                   // wave-uniform: EXEC stays all-1s

  const int n  = lane & 15;
  const int hi = lane >> 4;                       // 0: K pair {0,1}, 1: K pair {2,3}
  const long rowBase = (long)strip * 16;
  const float* Arow = A + (rowBase + n) * (long)K;

  v8f acc0 = {}, acc1 = {}, acc2 = {}, acc3 = {};
  for (int k = 0; k < K; k += 4) {
    const int ka = k + 2 * hi;
    float2 av = *(const float2*)(Arow + ka);
    if (RELU) { av.x = fmaxf(av.x, 0.f); av.y = fmaxf(av.y, 0.f); }
    v2f a; a[0] = av.x; a[1] = av.y;

    // pair-row index: ka>>1 = k/2 + hi
    const float2* bp = &BsP[((k >> 1) + hi) * BSTRIDE + n];
    const v2f bA = *(const v2f*)(bp + 0);
    const v2f bB = *(const v2f*)(bp + 16);
    const v2f bC = *(const v2f*)(bp + 32);
    const v2f bD = *(const v2f*)(bp + 48);

    acc0 = __builtin_amdgcn_wmma_f32_16x16x4_f32(false, a, false, bA, (short)0, acc0, false, false);
    acc1 = __builtin_amdgcn_wmma_f32_16x16x4_f32(false, a, false, bB, (short)0, acc1, false, false);
    acc2 = __builtin_amdgcn_wmma_f32_16x16x4_f32(false, a, false, bC, (short)0, acc2, false, false);
    acc3 = __builtin_amdgcn_wmma_f32_16x16x4_f32(false, a, false, bD, (short)0, acc3, false, false);
  }

  float* Cout = C + (rowBase + 8 * hi) * 64 + n;
#pragma unroll
  for (int r = 0; r < 8; ++r) {
    Cout[r * 64 + 0]  = acc0[r];
    Cout[r * 64 + 16] = acc1[r];
    Cout[r * 64 + 32] = acc2[r];
    Cout[r * 64 + 48] = acc3[r];
  }
}

// ---------------------------------------------------------------------------
// Edge scatter: acc[dst[e]][:] += w[e] * t[src[e]][:]
// One wave per edge; each lane handles a float2 feature pair:
// 2x global_atomic_add_f32 per lane.
// ---------------------------------------------------------------------------
__global__ void scatter_edges(const float* __restrict__ t,
                              const int* __restrict__ src,
                              const int* __restrict__ dst,
                              const float* __restrict__ w,
                              float* __restrict__ acc,
                              int E) {
  const long idx = (long)blockIdx.x * blockDim.x + threadIdx.x;
  const int e  = (int)(idx >> 5);
  const int pr = (int)(idx & 31);
  if (e >= E) return;
  const int s = src[e], d = dst[e];
  const float ww = w[e];
  const float2 v = *(const float2*)(t + (long)s * 64 + pr * 2);
  float* p = acc + (long)d * 64 + pr * 2;
  unsafeAtomicAdd(p,     v.x * ww);
  unsafeAtomicAdd(p + 1, v.y * ww);
}

// ---------------------------------------------------------------------------
// Decode: out[p][o] = sum_c Wlin[o][c] * concat(z[ps[p]], z[pd[p]])[c]
// ---------------------------------------------------------------------------
__global__ void decode_kernel(const float* __restrict__ z,
                              const int* __restrict__ pos,
                              const float* __restrict__ Wlin,
                              float* __restrict__ out,
                              int P) {
  __shared__ float wl[256];
  for (int i = threadIdx.x; i < 256; i += blockDim.x) wl[i] = Wlin[i];
  __syncthreads();

  const int p = blockIdx.x * blockDim.x + threadIdx.x;
  if (p >= P) return;
  const int s = pos[p];
  const int d = pos[P + p];
  const float4* zs = (const float4*)(z + (long)s * 64);
  const float4* zd = (const float4*)(z + (long)d * 64);

  float a0 = 0.f, a1 = 0.f;
#pragma unroll
  for (int q = 0; q < 16; ++q) {
    const float4 v = zs[q];
    a0 += v.x * wl[4*q]       + v.y * wl[4*q + 1]       + v.z * wl[4*q + 2]       + v.w * wl[4*q + 3];
    a1 += v.x * wl[128 + 4*q] + v.y * wl[128 + 4*q + 1] + v.z * wl[128 + 4*q + 2] + v.w * wl[128 + 4*q + 3];
  }
#pragma unroll
  for (int q = 0; q < 16; ++q) {
    const float4 v = zd[q];
    a0 += v.x * wl[64 + 4*q]  + v.y * wl[64 + 4*q + 1]  + v.z * wl[64 + 4*q + 2]  + v.w * wl[64 + 4*q + 3];
    a1 += v.x * wl[192 + 4*q] + v.y * wl[192 + 4*q + 1] + v.z * wl[192 + 4*q + 2] + v.w * wl[192 + 4*q + 3];
  }
  out[2 * p]     = a0;
  out[2 * p + 1] = a1;
}

// ---------------------------------------------------------------------------
extern "C" void kernel_launch(void* const* d_in, const int* in_sizes, int n_in,
                              void* d_out, int out_size, void* d_ws, size_t ws_size,
                              hipStream_t stream) {
  const float* x    = (const float*)d_in[0];
  const int*   ei   = (const int*)d_in[1];   // [2,E] row-major: src then dst
  const float* w    = (const float*)d_in[2];
  const int*   pos  = (const int*)d_in[3];   // [2,P]
  const float* W1   = (const float*)d_in[4]; // [128,64]
  const float* W2   = (const float*)d_in[5]; // [64,64]
  const float* Wlin = (const float*)d_in[6]; // [2,128]
  float* out = (float*)d_out;

  const int F = 128;
  const int N = in_sizes[0] / F;
  const int E = in_sizes[2];
  const int P = in_sizes[3] / 2;

  float* t1 = (float*)d_ws;                  // x @ W1            [N,64]
  float* h  = t1 + (size_t)N * 64;           // layer-1 aggregate [N,64]
  float* t2 = h  + (size_t)N * 64;           // relu(h) @ W2      [N,64]
  float* z  = t2 + (size_t)N * 64;           // layer-2 aggregate [N,64]

  // accumulator buffers must start from zero every call
  hipMemsetAsync(h, 0, (size_t)N * 64 * sizeof(float), stream);
  hipMemsetAsync(z, 0, (size_t)N * 64 * sizeof(float), stream);

  const int gemmBlocks = (N / 16 + 7) / 8;
  const long sthreads  = (long)E * 32;
  const int  sblocks   = (int)((sthreads + 255) / 256);

  gemm_wmma_f32<0><<<gemmBlocks, 256, 0, stream>>>(x, W1, t1, N, F);
  scatter_edges<<<sblocks, 256, 0, stream>>>(t1, ei, ei + E, w, h, E);
  gemm_wmma_f32<1><<<gemmBlocks, 256, 0, stream>>>(h, W2, t2, N, 64);
  scatter_edges<<<sblocks, 256, 0, stream>>>(t2, ei, ei + E, w, z, E);
  decode_kernel<<<(P + 255) / 256, 256, 0, stream>>>(z, pos, Wlin, out, P);
}